// HyperFNS_4509715660938
// MI455X (gfx1250) — compile-verified
//
// HyperFNS solver for MI455X (gfx1250, CDNA5, wave32).
// One workgroup (8 wave32) per sample; all per-sample state LDS-resident.
// The 63-point 2D DFTs run as dense 64x64 fp32 matmuls on V_WMMA_F32_16X16X4_F32.
// This revision preloads full K-sweeps of A/B fragments into VGPRs so the 16-step
// WMMA chains issue back-to-back (one s_wait_dscnt instead of one per WMMA).
#include <hip/hip_runtime.h>
#include <math.h>

#define NG    63
#define NP    64
#define SRT   65            // LDS row stride (64 banks -> conflict-free)
#define BATCH 256
#define PIX   (NG * NG)     // 3969
#define MID   32
#define ML    3
#define KS    7
#define K_ITERS 5

typedef float v2f __attribute__((ext_vector_type(2)));
typedef float v8f __attribute__((ext_vector_type(8)));

#if defined(__AMDGCN__)
#  if __has_builtin(__builtin_amdgcn_wmma_f32_16x16x4_f32)
#    define HAVE_WMMA_F32 1
#  endif
#endif
#ifndef HAVE_WMMA_F32
#  define HAVE_WMMA_F32 0
#endif

__device__ __forceinline__ float gelu_exact(float v) {
    return 0.5f * v * (1.0f + erff(v * 0.70710678118654752f));
}

// C[64x64] = A1*B1 (+ s2 * A2*B2), all LDS, row stride SRT. 256 threads = 8 waves,
// each wave owns two 16x16 C tiles (same column tile -> B fragments shared).
// Fragment layouts per CDNA5 WMMA fp32 spec:
//   A 16x4: lane L holds A[L%16][(L/16)*2 + {0,1}]
//   B 4x16: lane L holds B[(L/16)*2 + {0,1}][L%16]
//   C 16x16: vgpr v, lane L -> C[v + (L/16)*8][L%16]
template <bool DUAL>
__device__ __forceinline__ void mm64(const float* __restrict__ A1,
                                     const float* __restrict__ B1,
                                     const float* __restrict__ A2,
                                     const float* __restrict__ B2,
                                     const float s2,
                                     float* __restrict__ C)
{
    const int lane = threadIdx.x & 31;
    const int wave = threadIdx.x >> 5;
    const int half = lane >> 4;
    const int l16  = lane & 15;
#if HAVE_WMMA_F32
    const int tj = (wave & 3) << 4;             // column tile (same for both reps)
    // Preload B fragments once; both C tiles of this wave reuse them.
    v2f bf1[16], bf2[16];
#pragma unroll
    for (int kb = 0; kb < 16; ++kb) {
        const int k0 = kb * 4 + half * 2;
        bf1[kb].x = B1[k0 * SRT + tj + l16];
        bf1[kb].y = B1[(k0 + 1) * SRT + tj + l16];
        if (DUAL) {
            bf2[kb].x = B2[k0 * SRT + tj + l16];
            bf2[kb].y = B2[(k0 + 1) * SRT + tj + l16];
        }
    }
#pragma unroll
    for (int rep = 0; rep < 2; ++rep) {
        const int ti = ((wave >> 2) + rep * 2) << 4;   // row tile: wave and wave+8
        // Preload the whole K sweep of A fragments, then run the WMMA chain
        // back-to-back: a single LDS wait instead of one per WMMA step.
        v2f af1[16], af2[16];
#pragma unroll
        for (int kb = 0; kb < 16; ++kb) {
            const int k0 = kb * 4 + half * 2;
            af1[kb].x = A1[(ti + l16) * SRT + k0];
            af1[kb].y = A1[(ti + l16) * SRT + k0 + 1];
            if (DUAL) {
                af2[kb].x = A2[(ti + l16) * SRT + k0];
                af2[kb].y = A2[(ti + l16) * SRT + k0 + 1];
            }
        }
        v8f acc  = {0.f, 0.f, 0.f, 0.f, 0.f, 0.f, 0.f, 0.f};
        v8f accB = {0.f, 0.f, 0.f, 0.f, 0.f, 0.f, 0.f, 0.f};
        // Two independent accumulator chains (DUAL: product 1 / product 2;
        // single: even-k / odd-k) so consecutive WMMAs never RAW on one D.
#pragma unroll
        for (int kb = 0; kb < 16; kb += 2) {
            if (DUAL) {
                acc  = __builtin_amdgcn_wmma_f32_16x16x4_f32(false, af1[kb], false, bf1[kb],
                                                             (short)0, acc, false, false);
                accB = __builtin_amdgcn_wmma_f32_16x16x4_f32(false, af2[kb], false, bf2[kb],
                                                             (short)0, accB, false, false);
                acc  = __builtin_amdgcn_wmma_f32_16x16x4_f32(false, af1[kb + 1], false, bf1[kb + 1],
                                                             (short)0, acc, false, false);
                accB = __builtin_amdgcn_wmma_f32_16x16x4_f32(false, af2[kb + 1], false, bf2[kb + 1],
                                                             (short)0, accB, false, false);
            } else {
                acc  = __builtin_amdgcn_wmma_f32_16x16x4_f32(false, af1[kb], false, bf1[kb],
                                                             (short)0, acc, false, false);
                accB = __builtin_amdgcn_wmma_f32_16x16x4_f32(false, af1[kb + 1], false, bf1[kb + 1],
                                                             (short)0, accB, false, false);
            }
        }
#pragma unroll
        for (int v = 0; v < 8; ++v) {
            const float val = DUAL ? (acc[v] + s2 * accB[v]) : (acc[v] + accB[v]);
            C[(ti + half * 8 + v) * SRT + tj + l16] = val;
        }
    }
#else
    // Layout-agnostic LDS fallback (only used if the f32 WMMA builtin is absent).
    for (int tt = wave; tt < 16; tt += 8) {
        const int ti = (tt >> 2) << 4;
        const int tj = (tt & 3) << 4;
        for (int v = 0; v < 8; ++v) {
            const int row = ti + half * 8 + v;
            const int col = tj + l16;
            float acc = 0.f, acc2 = 0.f;
            for (int k = 0; k < NP; ++k) {
                acc += A1[row * SRT + k] * B1[k * SRT + col];
                if (DUAL) acc2 += A2[row * SRT + k] * B2[k * SRT + col];
            }
            C[row * SRT + col] = DUAL ? (acc + s2 * acc2) : acc;
        }
    }
#endif
}

// ---------------------------------------------------------------- utilities
__global__ void copy_kernel(const float* __restrict__ src, float* __restrict__ dst, int n)
{
    int i = blockIdx.x * blockDim.x + threadIdx.x;
    if (i < n) dst[i] = src[i];
}

// W[i][j] = exp(-2*pi*i*j/63) / sqrt(63), padded with zero row/col 63 so that the
// 64-wide WMMA K-loop reproduces the exact 63-point ortho DFT.
__global__ void dft_init_kernel(float* __restrict__ Wr, float* __restrict__ Wi)
{
    int p = blockIdx.x * blockDim.x + threadIdx.x;
    if (p >= NP * NP) return;
    int i = p >> 6, j = p & 63;
    if (i >= NG || j >= NG) { Wr[p] = 0.f; Wi[p] = 0.f; return; }
    const double ang = -2.0 * 3.14159265358979323846 * (double)((i * j) % NG) / (double)NG;
    const float sc = 0.1259881576697424f;  // 1/sqrt(63)
    Wr[p] = (float)cos(ang) * sc;
    Wi[p] = (float)sin(ang) * sc;
}

// ------------------------------------------------------------- hypernetwork
// Two MLPs 9 -> 100 (exact GELU) -> 147, one block per sample.
__global__ void hyper_kernel(const float* __restrict__ kA,
                             const float* __restrict__ w1a, const float* __restrict__ b1a,
                             const float* __restrict__ w2a, const float* __restrict__ b2a,
                             const float* __restrict__ w1b, const float* __restrict__ b1b,
                             const float* __restrict__ w2b, const float* __restrict__ b2b,
                             float* __restrict__ o1, float* __restrict__ o2)
{
    __shared__ float a9[9];
    __shared__ float h1[100];
    __shared__ float h2[100];
    const int b = blockIdx.x, tid = threadIdx.x;
    if (tid < 9) a9[tid] = kA[b * 9 + tid];
    __syncthreads();
    if (tid < 100) {
        float s1 = b1a[tid], s2 = b1b[tid];
        for (int k = 0; k < 9; ++k) {
            s1 += a9[k] * w1a[k * 100 + tid];
            s2 += a9[k] * w1b[k * 100 + tid];
        }
        h1[tid] = gelu_exact(s1);
        h2[tid] = gelu_exact(s2);
    }
    __syncthreads();
    if (tid < ML * KS * KS) {
        float s1 = b2a[tid], s2 = b2b[tid];
        for (int k = 0; k < 100; ++k) {
            s1 += h1[k] * w2a[k * 147 + tid];
            s2 += h2[k] * w2b[k * 147 + tid];
        }
        o1[b * 147 + tid] = s1;
        o2[b * 147 + tid] = s2;
    }
}

// ---------------------------------------------------- meta conv-transpose net
// torch ConvTranspose2d: out[oc][oy][ox] += in[ic][iy][ix] * w[ic][oc][ky][kx]
// with oy = iy*2 - pad + ky  =>  iy = (oy + pad - ky) / 2 when even & in range.
__device__ __forceinline__ void ctrans_stage(const float* __restrict__ in, int inC, int inW,
                                             float* __restrict__ out, int outC, int outW,
                                             const float* __restrict__ w,
                                             const float* __restrict__ bias,
                                             int pad, bool act, int tid)
{
    const int total = outC * outW * outW;
    for (int p = tid; p < total; p += 256) {
        const int oc = p / (outW * outW);
        const int rem = p % (outW * outW);
        const int oy = rem / outW, ox = rem % outW;
        float acc = bias[oc];
        for (int ky = 0; ky < 3; ++ky) {
            const int ty = oy + pad - ky;
            if (ty < 0 || (ty & 1) || (ty >> 1) >= inW) continue;
            const int iy = ty >> 1;
            for (int kx = 0; kx < 3; ++kx) {
                const int tx = ox + pad - kx;
                if (tx < 0 || (tx & 1) || (tx >> 1) >= inW) continue;
                const int ix = tx >> 1;
                for (int ic = 0; ic < inC; ++ic)
                    acc += in[(ic * inW + iy) * inW + ix] *
                           w[((ic * outC + oc) * 3 + ky) * 3 + kx];
            }
        }
        out[p] = act ? gelu_exact(acc) : acc;
    }
}

// Fully fused 5-layer chain per sample, all intermediates in LDS (max 139 KB stage).
// LDS partition: bufA 9248 f | bufB 34848 f | kbuf 16 f  => 176448 B dynamic.
__global__ void meta_kernel(const float* __restrict__ kA,
                            const float* __restrict__ w1, const float* __restrict__ b1,
                            const float* __restrict__ w2, const float* __restrict__ b2,
                            const float* __restrict__ w3, const float* __restrict__ b3,
                            const float* __restrict__ w4, const float* __restrict__ b4,
                            const float* __restrict__ w5, const float* __restrict__ b5,
                            float* __restrict__ hout)
{
    extern __shared__ float sm[];
    float* bufA = sm;             // 9248 floats
    float* bufB = sm + 9248;      // 34848 floats
    float* kbuf = bufB + 34848;   // 16 floats
    const int b = blockIdx.x, tid = threadIdx.x;
    if (tid < 9) kbuf[tid] = kA[b * 9 + tid];
    __syncthreads();
    ctrans_stage(kbuf, 1, 3, bufA, MID, 5, w1, b1, 1, true, tid);   __syncthreads();
    ctrans_stage(bufA, MID, 5, bufB, MID, 9, w2, b2, 1, true, tid); __syncthreads();
    ctrans_stage(bufB, MID, 9, bufA, MID, 17, w3, b3, 1, true, tid); __syncthreads();
    ctrans_stage(bufA, MID, 17, bufB, MID, 33, w4, b4, 1, true, tid); __syncthreads();
    ctrans_stage(bufB, MID, 33, bufA, 2, NG, w5, b5, 2, false, tid);  __syncthreads();
    for (int p = tid; p < 2 * PIX; p += 256) hout[b * 2 * PIX + p] = bufA[p];
}

// ------------------------------------------------------------------ residual
// pad_bc: top/left zero, bottom/right one; r = f - conv3x3(xpad, kA)
__global__ void residual_kernel(const float* __restrict__ xcur,
                                const float* __restrict__ f,
                                const float* __restrict__ kA,
                                float* __restrict__ r)
{
    __shared__ float k9[9];
    const int b = blockIdx.x, tid = threadIdx.x;
    if (tid < 9) k9[tid] = kA[b * 9 + tid];
    __syncthreads();
    const float* xs = xcur + b * PIX;
    for (int p = tid; p < PIX; p += 256) {
        const int y = p / NG, x = p % NG;
        float acc = 0.f;
        for (int i = 0; i < 3; ++i) {
            const int u = y + i;
            for (int j = 0; j < 3; ++j) {
                const int v = x + j;
                float val;
                if (u == NP || v == NP)      val = 1.f;
                else if (u == 0 || v == 0)   val = 0.f;
                else                         val = xs[(u - 1) * NG + (v - 1)];
                acc += val * k9[i * 3 + j];
            }
        }
        r[b * PIX + p] = f[b * PIX + p] - acc;
    }
}

// ------------------------------------------------------------------ smoother
// x += conv7x7(conv7x7(r, w1[3,1,7,7]), w2[1,3,7,7]), SAME zero pad.
// LDS: rpad 69*69 | mid 3*69*69 | k1 147 | k2 147 => 77352 B dynamic.
__global__ void smoother_kernel(const float* __restrict__ r,
                                const float* __restrict__ w1,
                                const float* __restrict__ w2,
                                float* __restrict__ xcur)
{
    extern __shared__ float sm[];
    const int PW = NG + 6;                // 69
    float* rp  = sm;                      // PW*PW
    float* mid = sm + PW * PW;            // 3*PW*PW
    float* k1  = mid + 3 * PW * PW;       // 147
    float* k2  = k1 + 147;                // 147
    const int b = blockIdx.x, tid = threadIdx.x;
    if (tid < 147) { k1[tid] = w1[b * 147 + tid]; k2[tid] = w2[b * 147 + tid]; }
    for (int p = tid; p < PW * PW; p += 256) {
        const int u = p / PW, v = p % PW;
        const bool in = (u >= 3 && u < NG + 3 && v >= 3 && v < NG + 3);
        rp[p] = in ? r[b * PIX + (u - 3) * NG + (v - 3)] : 0.f;
    }
    for (int p = tid; p < 3 * PW * PW; p += 256) mid[p] = 0.f;
    __syncthreads();
    for (int p = tid; p < ML * PIX; p += 256) {
        const int c = p / PIX, rem = p % PIX;
        const int y = rem / NG, x = rem % NG;
        float acc = 0.f;
        for (int i = 0; i < KS; ++i)
            for (int j = 0; j < KS; ++j)
                acc += rp[(y + i) * PW + (x + j)] * k1[c * 49 + i * KS + j];
        mid[c * PW * PW + (y + 3) * PW + (x + 3)] = acc;
    }
    __syncthreads();
    for (int p = tid; p < PIX; p += 256) {
        const int y = p / NG, x = p % NG;
        float acc = 0.f;
        for (int c = 0; c < ML; ++c)
            for (int i = 0; i < KS; ++i)
                for (int j = 0; j < KS; ++j)
                    acc += mid[c * PW * PW + (y + i) * PW + (x + j)] * k2[c * 49 + i * KS + j];
        xcur[b * PIX + p] += acc;
    }
}

// --------------------------------------------------------- spectral correction
// fft2_ortho(r) = W r W (W symmetric, 1/sqrt(63) folded in). Multiply by wc,
// keep cols 0..31, hermitian-rebuild cols 32..62, then real(conj(W) H conj(W)).
// 7 WMMA matmul passes, everything LDS-resident. x += y.
// LDS: 7 arrays of 64*65 floats = 116480 B dynamic.
__global__ void correct_kernel(float* __restrict__ xcur,
                               const float* __restrict__ r,
                               const float* __restrict__ h,
                               const float* __restrict__ gWr,
                               const float* __restrict__ gWi)
{
    extern __shared__ float sm[];
    float* Wr = sm;
    float* Wi = Wr + NP * SRT;
    float* Rb = Wi + NP * SRT;   // real input r / final y
    float* Tr = Rb + NP * SRT;   // T = W r   (then U)
    float* Ti = Tr + NP * SRT;
    float* Rr = Ti + NP * SRT;   // Rhat = T W (then O, then H)
    float* Ri = Rr + NP * SRT;
    const int b = blockIdx.x, tid = threadIdx.x;

    for (int p = tid; p < NP * NP; p += 256) {
        const int i = p >> 6, j = p & 63;
        Wr[i * SRT + j] = gWr[p];
        Wi[i * SRT + j] = gWi[p];
        Rb[i * SRT + j] = (i < NG && j < NG) ? r[b * PIX + i * NG + j] : 0.f;
    }
    __syncthreads();

    mm64<false>(Wr, Rb, nullptr, nullptr, 0.f, Tr);      // Tr = Wr r
    mm64<false>(Wi, Rb, nullptr, nullptr, 0.f, Ti);      // Ti = Wi r
    __syncthreads();
    mm64<true>(Tr, Wr, Ti, Wi, -1.f, Rr);                // Re(fft2)
    mm64<true>(Tr, Wi, Ti, Wr, +1.f, Ri);                // Im(fft2)
    __syncthreads();

    const float* hb = h + b * 2 * PIX;                   // [.., y, x, {re,im}] flat view
    for (int p = tid; p < PIX; p += 256) {
        const int y = p / NG, x = p % NG;
        const float wr = hb[y * 2 * NG + 2 * x];
        const float wi = hb[y * 2 * NG + 2 * x + 1];
        const float ar = Rr[y * SRT + x], ai = Ri[y * SRT + x];
        Rr[y * SRT + x] = ar * wr - ai * wi;
        Ri[y * SRT + x] = ar * wi + ai * wr;
    }
    __syncthreads();
    // hermitian rebuild of cols 32..62 from cols 1..31 (irfft2 of trimmed spectrum)
    for (int p = tid; p < NG * 31; p += 256) {
        const int k1 = p / 31;
        const int k2 = 32 + p % 31;
        const int sr = (NG - k1) % NG;
        const int sc = NG - k2;                          // 1..31
        Rr[k1 * SRT + k2] =  Rr[sr * SRT + sc];
        Ri[k1 * SRT + k2] = -Ri[sr * SRT + sc];
    }
    __syncthreads();

    mm64<true>(Wr, Rr, Wi, Ri, +1.f, Tr);                // Ur = Wr H_r + Wi H_i
    mm64<true>(Wr, Ri, Wi, Rr, -1.f, Ti);                // Ui = Wr H_i - Wi H_r
    __syncthreads();
    mm64<true>(Tr, Wr, Ti, Wi, +1.f, Rb);                // y = Ur Wr + Ui Wi
    __syncthreads();

    for (int p = tid; p < PIX; p += 256)
        xcur[b * PIX + p] += Rb[(p / NG) * SRT + (p % NG)];
}

// --------------------------------------------------------------- reductions
__global__ void reduce1_kernel(const float* __restrict__ r, float* __restrict__ partial)
{
    __shared__ float s[256];
    const int b = blockIdx.x, tid = threadIdx.x;
    float acc = 0.f;
    for (int p = tid; p < PIX; p += 256) {
        const float v = r[b * PIX + p];
        acc += v * v;
    }
    s[tid] = acc;
    __syncthreads();
    for (int ofs = 128; ofs > 0; ofs >>= 1) {
        if (tid < ofs) s[tid] += s[tid + ofs];
        __syncthreads();
    }
    if (tid == 0) partial[b] = s[0];
}

__global__ void reduce2_kernel(const float* __restrict__ partial, float* __restrict__ out)
{
    __shared__ float s[256];
    const int tid = threadIdx.x;
    s[tid] = partial[tid];
    __syncthreads();
    for (int ofs = 128; ofs > 0; ofs >>= 1) {
        if (tid < ofs) s[tid] += s[tid + ofs];
        __syncthreads();
    }
    if (tid == 0) out[0] = sqrtf(s[0]) / (float)BATCH;
}

// ------------------------------------------------------------------- launch
extern "C" void kernel_launch(void* const* d_in, const int* in_sizes, int n_in,
                              void* d_out, int out_size, void* d_ws, size_t ws_size,
                              hipStream_t stream)
{
    const float* x      = (const float*)d_in[0];
    const float* f      = (const float*)d_in[1];
    const float* kA     = (const float*)d_in[2];
    const float* fc1_w1 = (const float*)d_in[3];
    const float* fc1_b1 = (const float*)d_in[4];
    const float* fc1_w2 = (const float*)d_in[5];
    const float* fc1_b2 = (const float*)d_in[6];
    const float* fc2_w1 = (const float*)d_in[7];
    const float* fc2_b1 = (const float*)d_in[8];
    const float* fc2_w2 = (const float*)d_in[9];
    const float* fc2_b2 = (const float*)d_in[10];
    const float* ct1_w  = (const float*)d_in[11];
    const float* ct1_b  = (const float*)d_in[12];
    const float* ct2_w  = (const float*)d_in[13];
    const float* ct2_b  = (const float*)d_in[14];
    const float* ct3_w  = (const float*)d_in[15];
    const float* ct3_b  = (const float*)d_in[16];
    const float* ct4_w  = (const float*)d_in[17];
    const float* ct4_b  = (const float*)d_in[18];
    const float* ct5_w  = (const float*)d_in[19];
    const float* ct5_b  = (const float*)d_in[20];

    float* ws = (float*)d_ws;
    float* xcur    = ws;                    // BATCH*PIX
    float* rbuf    = xcur + BATCH * PIX;    // BATCH*PIX
    float* hbuf    = rbuf + BATCH * PIX;    // BATCH*2*PIX
    float* w1s     = hbuf + 2 * BATCH * PIX;// BATCH*147
    float* w2s     = w1s + BATCH * 147;     // BATCH*147
    float* Wr      = w2s + BATCH * 147;     // 4096
    float* Wi      = Wr + NP * NP;          // 4096
    float* partial = Wi + NP * NP;          // 256

    const int lds_meta = (9248 + 34848 + 16) * 4;            // 176448 B
    const int lds_sm   = (69 * 69 * 4 + 147 * 2) * 4;        // 77352 B
    const int lds_cor  = 7 * NP * SRT * 4;                   // 116480 B
    (void)hipFuncSetAttribute(reinterpret_cast<const void*>(meta_kernel),
                              hipFuncAttributeMaxDynamicSharedMemorySize, lds_meta);
    (void)hipFuncSetAttribute(reinterpret_cast<const void*>(smoother_kernel),
                              hipFuncAttributeMaxDynamicSharedMemorySize, lds_sm);
    (void)hipFuncSetAttribute(reinterpret_cast<const void*>(correct_kernel),
                              hipFuncAttributeMaxDynamicSharedMemorySize, lds_cor);

    copy_kernel<<<(BATCH * PIX + 255) / 256, 256, 0, stream>>>(x, xcur, BATCH * PIX);
    dft_init_kernel<<<(NP * NP + 255) / 256, 256, 0, stream>>>(Wr, Wi);
    hyper_kernel<<<BATCH, 256, 0, stream>>>(kA, fc1_w1, fc1_b1, fc1_w2, fc1_b2,
                                            fc2_w1, fc2_b1, fc2_w2, fc2_b2, w1s, w2s);
    meta_kernel<<<BATCH, 256, lds_meta, stream>>>(kA, ct1_w, ct1_b, ct2_w, ct2_b,
                                                  ct3_w, ct3_b, ct4_w, ct4_b,
                                                  ct5_w, ct5_b, hbuf);
    for (int it = 0; it < K_ITERS; ++it) {
        residual_kernel<<<BATCH, 256, 0, stream>>>(xcur, f, kA, rbuf);
        smoother_kernel<<<BATCH, 256, lds_sm, stream>>>(rbuf, w1s, w2s, xcur);
        residual_kernel<<<BATCH, 256, 0, stream>>>(xcur, f, kA, rbuf);
        correct_kernel<<<BATCH, 256, lds_cor, stream>>>(xcur, rbuf, hbuf, Wr, Wi);
    }
    residual_kernel<<<BATCH, 256, 0, stream>>>(xcur, f, kA, rbuf);
    reduce1_kernel<<<BATCH, 256, 0, stream>>>(rbuf, partial);
    reduce2_kernel<<<1, 256, 0, stream>>>(partial, (float*)d_out);
}